// CVXProjLoss_32014686224511
// MI455X (gfx1250) — compile-verified
//
#include <hip/hip_runtime.h>
#include <hip/hip_bf16.h>

// Problem constants (from reference)
#define BB 16
#define CC 1000
#define CP 1024   // padded constraint count (pad rows have w=0, G=0)
#define FF 256
#define KK 10
#define IPM_ITERS 20
#define SIGMA_C 0.1f
#define BIG_C 1e10f

typedef __attribute__((ext_vector_type(2))) float v2f;
typedef __attribute__((ext_vector_type(8))) float v8f;

// ---------------------------------------------------------------------------
// Kernel 0: build G = -D@W (sparse-D closed form), h, init z/lam/s/w.
//   partner(b,c): t_last by default; targets[j] -> targets[j-1]; targets[0] -> zero row.
// ---------------------------------------------------------------------------
__global__ void prep_kernel(const float* __restrict__ feats,
                            const int* __restrict__ tgt,
                            const float* __restrict__ W,
                            const float* __restrict__ bias,
                            const float* __restrict__ margin,
                            float* __restrict__ G, float* __restrict__ h,
                            float* __restrict__ lam, float* __restrict__ s,
                            float* __restrict__ w, float* __restrict__ z) {
  const int b = blockIdx.x;
  const int t = threadIdx.x; // 256 threads, t == feature index
  __shared__ int tg[KK];
  if (t < KK) tg[t] = tgt[b * KK + t];
  __syncthreads();
  const float mg = margin[0];

  for (int r = 0; r < CP; ++r) {
    float g = 0.f, hv = 0.f;
    if (r < CC) {
      int j = -1;
#pragma unroll
      for (int q = 0; q < KK; ++q)
        if (tg[q] == r) j = q;
      if (j != 0) {  // j==0: excluded row (zero G row, h=0)
        const int p = (j > 0) ? tg[j - 1] : tg[KK - 1];
        g = W[r * FF + t] - W[p * FF + t];
        hv = bias[p] - bias[r] - mg;
      }
    }
    G[((size_t)b * CP + r) * FF + t] = g;
    if (t == 0) h[b * CP + r] = hv;
  }
  for (int r = t; r < CP; r += 256) {
    lam[b * CP + r] = (r < CC) ? 1.f : 0.f;
    s[b * CP + r] = 1.f;
    w[b * CP + r] = 0.f;   // pad rows stay 0 forever -> no Gram contribution
  }
  if (t < FF) z[b * FF + t] = feats[b * FF + t]; // z0 = -0.5*p = feats
}

// ---------------------------------------------------------------------------
// Kernel 1 (per iteration): rp, mu, rc, w, and rhs = -(2z + p + G^T(lam + (lam*rp-rc)/s))
// One block (8 waves) per batch. Wave-per-row GEMV with shfl reductions.
// ---------------------------------------------------------------------------
__global__ void resid_kernel(const float* __restrict__ feats,
                             const float* __restrict__ G,
                             const float* __restrict__ h,
                             const float* __restrict__ lam_g,
                             const float* __restrict__ s_g,
                             const float* __restrict__ z_g,
                             float* __restrict__ w_g, float* __restrict__ rp_g,
                             float* __restrict__ rc_g, float* __restrict__ rhs_g) {
  const int b = blockIdx.x;
  const int t = threadIdx.x;
  const int lane = t & 31, wv = t >> 5;
  __shared__ float zl[FF];
  __shared__ float coef[CP];
  __shared__ float rp_sh[CP];
  __shared__ float red[8];
  if (t < FF) zl[t] = z_g[b * FF + t];
  __syncthreads();
  const float* Gb = G + (size_t)b * CP * FF;

  float zreg[8];
#pragma unroll
  for (int j = 0; j < 8; ++j) zreg[j] = zl[lane + 32 * j];

  // Phase A: rp_r = G[r,:].z + s_r - h_r ; accumulate lam*s
  float musum = 0.f;
  for (int i = 0; i < 125; ++i) {
    const int r = wv + 8 * i;
    const float* Gr = Gb + (size_t)r * FF;
    float acc = 0.f;
#pragma unroll
    for (int j = 0; j < 8; ++j) acc += Gr[lane + 32 * j] * zreg[j];
#pragma unroll
    for (int o = 16; o > 0; o >>= 1) acc += __shfl_xor(acc, o, 32);
    if (lane == 0) {
      const float sv = s_g[b * CP + r];
      const float lv = lam_g[b * CP + r];
      const float rp = acc + sv - h[b * CP + r];
      rp_sh[r] = rp;
      rp_g[b * CP + r] = rp;
      musum += lv * sv;
    }
  }
  if (lane == 0) red[wv] = musum;
  __syncthreads();
  float mu = 0.f;
#pragma unroll
  for (int q = 0; q < 8; ++q) mu += red[q];
  mu *= (1.f / (float)CC);

  // Phase B: rc, w, coef = lam + (lam*rp - rc)/s
  for (int r = t; r < CC; r += 256) {
    const float lv = lam_g[b * CP + r];
    const float sv = s_g[b * CP + r];
    const float rc = lv * sv - SIGMA_C * mu;
    w_g[b * CP + r] = lv / sv;
    rc_g[b * CP + r] = rc;
    coef[r] = lv + (lv * rp_sh[r] - rc) / sv;
  }
  __syncthreads();

  // Phase C: rhs_f = -(2 z_f + p_f + sum_r G[r,f]*coef_r), coalesced over f=t
  {
    float acc = 0.f;
    for (int r = 0; r < CC; ++r) acc += Gb[(size_t)r * FF + t] * coef[r];
    const float pf = -2.f * feats[b * FF + t];
    rhs_g[b * FF + t] = -(2.f * zl[t] + pf + acc);
  }
}

// ---------------------------------------------------------------------------
// Kernel 2 (per iteration): M = 2I + G^T diag(w) G via V_WMMA_F32_16X16X4_F32.
// Grid = B*16 blocks; each block (8 waves) computes one 16-row block of M
// (16 tiles of 16x16; 2 tiles/wave). K-panel of G is transposed into LDS
// (stride 66 -> 8B-aligned b64 fragment loads); w folded into A-operand.
// ---------------------------------------------------------------------------
#define KC 64
#define LDT 66
__global__ void gram_kernel(const float* __restrict__ G,
                            const float* __restrict__ w_g,
                            float* __restrict__ M) {
  const int b = blockIdx.x >> 4;
  const int ti = blockIdx.x & 15;
  const int t = threadIdx.x;
  const int lane = t & 31, wv = t >> 5;
  extern __shared__ float smem[];
  float* Gt = smem;              // 256 * 66 floats (transposed K-panel)
  float* wl = smem + 256 * LDT;  // 64 floats (w chunk), 8B aligned

  const float* Gb = G + (size_t)b * CP * FF;
  const int m = lane & 15;
  const int hi = lane >> 4;      // 0: K pair {0,1}, 1: K pair {2,3}
  const int tj0 = wv * 2, tj1 = wv * 2 + 1;

  v8f acc0 = {};
  v8f acc1 = {};

  for (int k0 = 0; k0 < CP; k0 += KC) {
    __syncthreads();  // protect Gt reuse across chunks
    // cooperative load + transpose: Gt[f][kk] = G[k0+kk][f]
#pragma unroll
    for (int j = 0; j < 16; ++j) {
      const int q = t + 256 * j;        // float4 index in 64x256 panel
      const int kk = q >> 6;
      const int fb = (q & 63) * 4;
      const float4 val = *(const float4*)(Gb + (size_t)(k0 + kk) * FF + fb);
      Gt[(fb + 0) * LDT + kk] = val.x;
      Gt[(fb + 1) * LDT + kk] = val.y;
      Gt[(fb + 2) * LDT + kk] = val.z;
      Gt[(fb + 3) * LDT + kk] = val.w;
    }
    if (t < KC) wl[t] = w_g[b * CP + k0 + t];
    __syncthreads();

#pragma unroll
    for (int kk0 = 0; kk0 < KC; kk0 += 4) {
      const int kkA = kk0 + 2 * hi;
      v2f wp = *(const v2f*)(wl + kkA);
      v2f a = *(const v2f*)(Gt + (16 * ti + m) * LDT + kkA);
      a.x *= wp.x;
      a.y *= wp.y;
      const v2f b0 = *(const v2f*)(Gt + (16 * tj0 + m) * LDT + kkA);
      const v2f b1 = *(const v2f*)(Gt + (16 * tj1 + m) * LDT + kkA);
      acc0 = __builtin_amdgcn_wmma_f32_16x16x4_f32(false, a, false, b0,
                                                   (short)0, acc0, false, false);
      acc1 = __builtin_amdgcn_wmma_f32_16x16x4_f32(false, a, false, b1,
                                                   (short)0, acc1, false, false);
    }
  }

  // store D tiles; C/D layout: VGPR v -> row v (+8 for lanes 16-31), col = lane&15
  float* Mb = M + (size_t)b * FF * FF;
#pragma unroll
  for (int v = 0; v < 8; ++v) {
    const int row = 16 * ti + v + 8 * hi;
    const int c0 = 16 * tj0 + m;
    const int c1 = 16 * tj1 + m;
    Mb[row * FF + c0] = acc0[v] + ((row == c0) ? 2.f : 0.f);
    Mb[row * FF + c1] = acc1[v] + ((row == c1) ? 2.f : 0.f);
  }
}

// ---------------------------------------------------------------------------
// Kernel 3 (per iteration): in-LDS Gauss-Jordan solve of M dz = rhs (256x257
// padded matrix fully resident in CDNA5's 320KB LDS), then ds/dlam, alpha line
// search and state update.
// ---------------------------------------------------------------------------
__global__ void solve_update_kernel(const float* __restrict__ G,
                                    const float* __restrict__ rhs_g,
                                    const float* __restrict__ rp_g,
                                    const float* __restrict__ rc_g,
                                    const float* __restrict__ M,
                                    float* __restrict__ lam_g,
                                    float* __restrict__ s_g,
                                    float* __restrict__ z_g) {
  const int b = blockIdx.x;
  const int t = threadIdx.x;  // 1024 threads
  extern __shared__ float smem[];
  const int MS = FF + 1;      // 257: odd stride -> conflict-free column walks
  float* Ml = smem;                 // 256*257
  float* rv = Ml + FF * MS;         // 256
  float* dz = rv + FF;              // 256
  float* dsA = dz + FF;             // 1024
  float* dlA = dsA + CP;            // 1024
  float* red = dlA + CP;            // 32
  float* alpha_sh = red + 32;       // 1

  const float* Mb = M + (size_t)b * FF * FF;
  for (int q = t; q < FF * FF; q += 1024) Ml[(q >> 8) * MS + (q & 255)] = Mb[q];
  if (t < FF) rv[t] = rhs_g[b * FF + t];
  __syncthreads();

  // Gauss-Jordan (SPD, no pivoting). Skip pivot column writes in-step to stay
  // race-free; stale sub-pivot columns are never read again.
  const int j = t & 255;
  const int rg = t >> 8;  // 0..3
  for (int k = 0; k < FF; ++k) {
    const float invp = 1.f / Ml[k * MS + k];
    const float mkj = Ml[k * MS + j];
    const float rk = rv[k];
    for (int i = rg; i < FF; i += 4) {
      if (i != k) {
        const float f = Ml[i * MS + k] * invp;
        if (j != k) Ml[i * MS + j] -= f * mkj;
        if (j == 0) rv[i] -= f * rk;
      }
    }
    __syncthreads();
  }
  if (t < FF) dz[t] = rv[t] / Ml[t * MS + t];
  __syncthreads();

  // ds = -rp - G dz ; dlam = -(rc + lam*ds)/s ; alpha line search
  const int lane = t & 31, wv = t >> 5;  // 32 waves
  const float* Gb = G + (size_t)b * CP * FF;
  float dzreg[8];
#pragma unroll
  for (int q = 0; q < 8; ++q) dzreg[q] = dz[lane + 32 * q];

  float amin = BIG_C;
  for (int i = 0; i < 32; ++i) {
    const int r = wv + 32 * i;
    if (r < CC) {
      const float* Gr = Gb + (size_t)r * FF;
      float acc = 0.f;
#pragma unroll
      for (int q = 0; q < 8; ++q) acc += Gr[lane + 32 * q] * dzreg[q];
#pragma unroll
      for (int o = 16; o > 0; o >>= 1) acc += __shfl_xor(acc, o, 32);
      if (lane == 0) {
        const float rp = rp_g[b * CP + r];
        const float rc = rc_g[b * CP + r];
        const float lv = lam_g[b * CP + r];
        const float sv = s_g[b * CP + r];
        const float ds = -rp - acc;
        const float dl = -(rc + lv * ds) / sv;
        dsA[r] = ds;
        dlA[r] = dl;
        if (ds < 0.f) amin = fminf(amin, -sv / ds);
        if (dl < 0.f) amin = fminf(amin, -lv / dl);
      }
    }
  }
  if (lane == 0) red[wv] = amin;
  __syncthreads();
  if (t == 0) {
    float a = BIG_C;
#pragma unroll
    for (int q = 0; q < 32; ++q) a = fminf(a, red[q]);
    alpha_sh[0] = fminf(1.f, 0.99f * a);
  }
  __syncthreads();
  const float alpha = alpha_sh[0];
  if (t < FF) z_g[b * FF + t] += alpha * dz[t];
  for (int r = t; r < CC; r += 1024) {
    lam_g[b * CP + r] += alpha * dlA[r];
    s_g[b * CP + r] += alpha * dsA[r];
  }
}

// ---------------------------------------------------------------------------
// Final: loss[b] = sum_f (feats - z)^2
// ---------------------------------------------------------------------------
__global__ void loss_kernel(const float* __restrict__ feats,
                            const float* __restrict__ z_g,
                            float* __restrict__ out) {
  const int b = blockIdx.x;
  const int t = threadIdx.x;  // 256
  __shared__ float red[256];
  const float d = feats[b * FF + t] - z_g[b * FF + t];
  red[t] = d * d;
  __syncthreads();
  for (int o = 128; o > 0; o >>= 1) {
    if (t < o) red[t] += red[t + o];
    __syncthreads();
  }
  if (t == 0) out[b] = red[0];
}

// ---------------------------------------------------------------------------
extern "C" void kernel_launch(void* const* d_in, const int* in_sizes, int n_in,
                              void* d_out, int out_size, void* d_ws, size_t ws_size,
                              hipStream_t stream) {
  (void)in_sizes; (void)n_in; (void)out_size; (void)ws_size;
  // inputs (setup_inputs order): logits_pred(unused), feats_pred, feats_pred_0(unused),
  // attack_targets, head_W, head_bias, margin
  const float* feats = (const float*)d_in[1];
  const int* tgt = (const int*)d_in[3];
  const float* W = (const float*)d_in[4];
  const float* bias = (const float*)d_in[5];
  const float* margin = (const float*)d_in[6];
  float* out = (float*)d_out;

  // workspace layout (floats), total ~21.4 MB
  float* ws = (float*)d_ws;
  size_t off = 0;
  float* G = ws + off;   off += (size_t)BB * CP * FF;  // 16.78 MB
  float* M = ws + off;   off += (size_t)BB * FF * FF;  // 4 MB
  float* h = ws + off;   off += (size_t)BB * CP;
  float* lam = ws + off; off += (size_t)BB * CP;
  float* s = ws + off;   off += (size_t)BB * CP;
  float* w = ws + off;   off += (size_t)BB * CP;
  float* rp = ws + off;  off += (size_t)BB * CP;
  float* rc = ws + off;  off += (size_t)BB * CP;
  float* z = ws + off;   off += (size_t)BB * FF;
  float* rhs = ws + off; off += (size_t)BB * FF;

  const size_t smem_gram = (size_t)(256 * LDT + 64) * sizeof(float);       // ~67.8 KB
  const size_t smem_solve =
      (size_t)(FF * (FF + 1) + FF + FF + CP + CP + 32 + 1) * sizeof(float); // ~273 KB

  prep_kernel<<<BB, 256, 0, stream>>>(feats, tgt, W, bias, margin, G, h, lam, s, w, z);
  for (int it = 0; it < IPM_ITERS; ++it) {
    resid_kernel<<<BB, 256, 0, stream>>>(feats, G, h, lam, s, z, w, rp, rc, rhs);
    gram_kernel<<<BB * 16, 256, smem_gram, stream>>>(G, w, M);
    solve_update_kernel<<<BB, 1024, smem_solve, stream>>>(G, rhs, rp, rc, M, lam, s, z);
  }
  loss_kernel<<<BB, 256, 0, stream>>>(feats, z, out);
}